// HETE_GCN_Optimized_67053029425732
// MI455X (gfx1250) — compile-verified
//
#include <hip/hip_runtime.h>
#include <hip/hip_bf16.h>

typedef float v2f __attribute__((ext_vector_type(2)));
typedef float v8f __attribute__((ext_vector_type(8)));

// ---------------------------------------------------------------------------
// Degree / normalization prep
// ---------------------------------------------------------------------------
__global__ void deg_init_k(float* __restrict__ deg, int N) {
    int i = blockIdx.x * blockDim.x + threadIdx.x;
    if (i < N) deg[i] = 1.0f;               // self-loop contributes 1
}

__global__ void deg_acc_k(const int* __restrict__ dst, const float* __restrict__ ew,
                          float* __restrict__ deg, int E) {
    int e = blockIdx.x * blockDim.x + threadIdx.x;
    if (e < E) atomicAdd(&deg[dst[e]], ew[e]);
}

__global__ void dinv_k(const float* __restrict__ deg, float* __restrict__ dinv, int N) {
    int i = blockIdx.x * blockDim.x + threadIdx.x;
    if (i < N) dinv[i] = rsqrtf(deg[i]);
}

__global__ void coef_k(const int* __restrict__ src, const int* __restrict__ dst,
                       const float* __restrict__ ew, const float* __restrict__ dinv,
                       float* __restrict__ coef, int E) {
    int e = blockIdx.x * blockDim.x + threadIdx.x;
    if (e < E) coef[e] = dinv[src[e]] * ew[e] * dinv[dst[e]];
}

// ---------------------------------------------------------------------------
// fp32 GEMM via V_WMMA_F32_16X16X4_F32.
// One wave computes a 16 x (NT*16) output tile; K stepped by 4.
// A layout: lanes 0-15 -> {K0,K1}, lanes 16-31 -> {K2,K3}, M = lane&15.
// B layout (K x NCOLS, row-major): symmetric, N = lane&15.
// C/D: VGPR r holds M=r (lanes 0-15) / M=r+8 (lanes 16-31), N = lane&15.
// A-row index is clamped (not predicated) so A loads stay unconditional
// b64 loads and EXEC remains all-ones (WMMA requirement). The store
// epilogue branches on a wave-uniform "full tile" test so the common
// path is a straight run of unguarded global_store_b32.
// ---------------------------------------------------------------------------
template <int K, int NT>
__global__ __launch_bounds__(256) void gemm_wmma_k(const float* __restrict__ A,
                                                   const float* __restrict__ B,
                                                   float* __restrict__ C, int M) {
    const int NCOLS = NT * 16;
    int wave = (int)((blockIdx.x * blockDim.x + threadIdx.x) >> 5);
    int lane = threadIdx.x & 31;
    int m0 = wave * 16;
    if (m0 >= M) return;                         // whole-wave exit: EXEC stays all-ones

    int mr  = lane & 15;
    int hi  = lane >> 4;
    int klo = hi << 1;                           // 0 or 2
    int row = m0 + mr;
    if (row >= M) row = M - 1;                   // clamp: keep loads unconditional

    v8f acc[NT];
#pragma unroll
    for (int n = 0; n < NT; ++n) acc[n] = (v8f){0.f, 0.f, 0.f, 0.f, 0.f, 0.f, 0.f, 0.f};

    const float* arow = A + (long)row * K + klo;

    for (int k0 = 0; k0 < K; k0 += 4) {
        v2f a = *(const v2f*)(arow + k0);        // 8B-aligned contiguous pair
#pragma unroll
        for (int n = 0; n < NT; ++n) {
            int col = n * 16 + mr;
            v2f b;
            b.x = B[(long)(k0 + klo) * NCOLS + col];
            b.y = B[(long)(k0 + klo + 1) * NCOLS + col];
            acc[n] = __builtin_amdgcn_wmma_f32_16x16x4_f32(
                false, a, false, b, (short)0, acc[n], false, false);
        }
    }

    float* crow = C + (long)(m0 + 8 * hi) * NCOLS + mr;   // row m0+8*hi, col mr
    if (m0 + 16 <= M) {
        // full tile (wave-uniform): unguarded store clause
#pragma unroll
        for (int n = 0; n < NT; ++n)
#pragma unroll
            for (int r = 0; r < 8; ++r)
                crow[(long)r * NCOLS + n * 16] = acc[n][r];
    } else {
#pragma unroll
        for (int n = 0; n < NT; ++n)
#pragma unroll
            for (int r = 0; r < 8; ++r)
                if (m0 + r + 8 * hi < M)
                    crow[(long)r * NCOLS + n * 16] = acc[n][r];
    }
}

// ---------------------------------------------------------------------------
// Aggregation: self-loop term, then one wave per edge scatter-add
// ---------------------------------------------------------------------------
template <int H>
__global__ void selfterm_k(const float* __restrict__ xw, const float* __restrict__ deg,
                           float* __restrict__ agg, int N) {
    long i = (long)blockIdx.x * blockDim.x + threadIdx.x;
    if (i >= (long)N * H) return;
    int node = (int)(i / H);
    agg[i] = xw[i] / deg[node];
}

template <int H>
__global__ __launch_bounds__(256) void scatter_k(const float* __restrict__ xw,
                                                 const int* __restrict__ src,
                                                 const int* __restrict__ dst,
                                                 const float* __restrict__ coef,
                                                 float* __restrict__ agg, int E) {
    long gid = (long)blockIdx.x * blockDim.x + threadIdx.x;
    int e    = (int)(gid >> 5);
    int lane = (int)(gid & 31);
    if (e >= E) return;
    int s = src[e];
    int d = dst[e];
    float c = coef[e];
#pragma unroll
    for (int f = lane; f < H; f += 32)
        atomicAdd(&agg[(long)d * H + f], xw[(long)s * H + f] * c);
}

// ---------------------------------------------------------------------------
// BatchNorm (training-mode batch stats) + ReLU
// ---------------------------------------------------------------------------
__global__ void zero_k(float* __restrict__ p, int n) {
    int i = blockIdx.x * blockDim.x + threadIdx.x;
    if (i < n) p[i] = 0.0f;
}

template <int C>
__global__ __launch_bounds__(256) void bn_stats_k(const float* __restrict__ h, int N,
                                                  float* __restrict__ stats /*2*C*/) {
    // 256 % C == 0 and stride is a multiple of C -> each thread owns one column
    int col = threadIdx.x % C;
    float s = 0.f, s2 = 0.f;
    long total  = (long)N * C;
    long stride = (long)gridDim.x * blockDim.x;
    for (long i = (long)blockIdx.x * blockDim.x + threadIdx.x; i < total; i += stride) {
        float v = h[i];
        s += v;
        s2 += v * v;
    }
    __shared__ float ls[256];
    __shared__ float ls2[256];
    ls[threadIdx.x]  = s;
    ls2[threadIdx.x] = s2;
    __syncthreads();
    if (threadIdx.x < C) {
        float a = 0.f, b = 0.f;
        for (int t = threadIdx.x; t < 256; t += C) { a += ls[t]; b += ls2[t]; }
        atomicAdd(&stats[col], a);
        atomicAdd(&stats[C + col], b);
    }
}

template <int C>
__global__ void bn_relu_k(const float* __restrict__ h, const float* __restrict__ stats,
                          const float* __restrict__ gamma, const float* __restrict__ beta,
                          float* __restrict__ out, int N) {
    long i = (long)blockIdx.x * blockDim.x + threadIdx.x;
    if (i >= (long)N * C) return;
    int c      = (int)(i % C);
    float invN = 1.0f / (float)N;
    float mean = stats[c] * invN;
    float var  = stats[C + c] * invN - mean * mean;
    float scl  = rsqrtf(var + 1e-5f) * gamma[c];
    float v    = (h[i] - mean) * scl + beta[c];
    out[i] = v > 0.f ? v : 0.f;
}

// ---------------------------------------------------------------------------
// Final linear [N,32] @ [32,1] + bl
// ---------------------------------------------------------------------------
__global__ void final_k(const float* __restrict__ h2, const float* __restrict__ Wl,
                        const float* __restrict__ bl, float* __restrict__ out, int N) {
    int i = blockIdx.x * blockDim.x + threadIdx.x;
    if (i >= N) return;
    const float* r = h2 + (long)i * 32;
    float s = bl[0];
#pragma unroll
    for (int c = 0; c < 32; ++c) s += r[c] * Wl[c];
    out[i] = s;
}

// ---------------------------------------------------------------------------
// Launch
// ---------------------------------------------------------------------------
extern "C" void kernel_launch(void* const* d_in, const int* in_sizes, int n_in,
                              void* d_out, int out_size, void* d_ws, size_t ws_size,
                              hipStream_t stream) {
    const float* x    = (const float*)d_in[0];
    const int*   ei   = (const int*)d_in[1];     // [2, E] flat
    const float* ew   = (const float*)d_in[2];
    const float* W1   = (const float*)d_in[3];
    const float* g1   = (const float*)d_in[5];
    const float* be1  = (const float*)d_in[6];
    const float* W2   = (const float*)d_in[7];
    const float* g2   = (const float*)d_in[9];
    const float* be2  = (const float*)d_in[10];
    const float* Wl   = (const float*)d_in[11];
    const float* bl   = (const float*)d_in[12];
    float*       out  = (float*)d_out;

    const int FIN = 128, H1 = 64, H2 = 32;
    const int N = in_sizes[0] / FIN;             // 100000
    const int E = in_sizes[2];                   // 1600000
    const int* src = ei;
    const int* dst = ei + E;

    // Workspace carve-out (256B aligned)
    char* ws = (char*)d_ws;
    size_t off = 0;
    auto carve = [&](size_t bytes) -> char* {
        char* p = ws + off;
        off = (off + bytes + 255) & ~(size_t)255;
        return p;
    };
    float* deg   = (float*)carve((size_t)N * 4);
    float* dinv  = (float*)carve((size_t)N * 4);
    float* coef  = (float*)carve((size_t)E * 4);
    float* buf1  = (float*)carve((size_t)N * H1 * 4);   // xw1, later h1
    float* agg1  = (float*)carve((size_t)N * H1 * 4);
    float* buf2  = (float*)carve((size_t)N * H2 * 4);   // xw2, later h2
    float* agg2  = (float*)carve((size_t)N * H2 * 4);
    float* stats = (float*)carve(2 * 64 * 4);
    (void)ws_size; (void)n_in; (void)out_size;

    const int T = 256;
    auto cdiv = [](long a, long b) { return (int)((a + b - 1) / b); };

    // --- degree / normalization coefficients ---
    deg_init_k<<<cdiv(N, T), T, 0, stream>>>(deg, N);
    deg_acc_k<<<cdiv(E, T), T, 0, stream>>>(dst, ew, deg, E);
    dinv_k<<<cdiv(N, T), T, 0, stream>>>(deg, dinv, N);
    coef_k<<<cdiv(E, T), T, 0, stream>>>(src, dst, ew, dinv, coef, E);

    // --- layer 1: xw1 = x @ W1 (WMMA f32), aggregate, BN+ReLU ---
    {
        int tiles = cdiv(N, 16);
        gemm_wmma_k<128, 4><<<cdiv((long)tiles * 32, T), T, 0, stream>>>(x, W1, buf1, N);
    }
    selfterm_k<H1><<<cdiv((long)N * H1, T), T, 0, stream>>>(buf1, deg, agg1, N);
    scatter_k<H1><<<cdiv((long)E * 32, T), T, 0, stream>>>(buf1, src, dst, coef, agg1, E);
    zero_k<<<1, 128, 0, stream>>>(stats, 2 * H1);
    bn_stats_k<H1><<<1024, T, 0, stream>>>(agg1, N, stats);
    bn_relu_k<H1><<<cdiv((long)N * H1, T), T, 0, stream>>>(agg1, stats, g1, be1, buf1, N);

    // --- layer 2: xw2 = h1 @ W2 (WMMA f32), aggregate, BN+ReLU ---
    {
        int tiles = cdiv(N, 16);
        gemm_wmma_k<64, 2><<<cdiv((long)tiles * 32, T), T, 0, stream>>>(buf1, W2, buf2, N);
    }
    selfterm_k<H2><<<cdiv((long)N * H2, T), T, 0, stream>>>(buf2, deg, agg2, N);
    scatter_k<H2><<<cdiv((long)E * 32, T), T, 0, stream>>>(buf2, src, dst, coef, agg2, E);
    zero_k<<<1, 128, 0, stream>>>(stats, 2 * H2);
    bn_stats_k<H2><<<1024, T, 0, stream>>>(agg2, N, stats);
    bn_relu_k<H2><<<cdiv((long)N * H2, T), T, 0, stream>>>(agg2, stats, g2, be2, buf2, N);

    // --- final linear ---
    final_k<<<cdiv(N, T), T, 0, stream>>>(buf2, Wl, bl, out, N);
}